// StructureAwareLoss_17875653886217
// MI455X (gfx1250) — compile-verified
//
#include <hip/hip_runtime.h>
#include <hip/hip_bf16.h>
#include <math.h>

#define VOCAB  32000
#define NROWS  4096
#define SEQ    1024
#define NBATCH 4

typedef float v2f __attribute__((ext_vector_type(2)));
typedef float v8f __attribute__((ext_vector_type(8)));
typedef int   v4i __attribute__((ext_vector_type(4)));

// ---- CDNA5 async-to-LDS path (probe-guarded) -------------------------------
#if defined(__AMDGCN__) && __has_builtin(__builtin_amdgcn_global_load_async_to_lds_b128)
#define USE_ASYNC 1
#else
#define USE_ASYNC 0
#endif

#if USE_ASYNC
#if __has_builtin(__builtin_amdgcn_s_wait_asynccnt)
#define WAIT_ASYNC(n) __builtin_amdgcn_s_wait_asynccnt(n)
#else
#define WAIT_ASYNC(n) asm volatile("s_wait_asynccnt %0" ::"n"(n) : "memory")
#endif
// global (AS1) and LDS (AS3) int4 pointers, per the builtin's signature
typedef __attribute__((address_space(1))) v4i* gas_p;
typedef __attribute__((address_space(3))) v4i* las_p;
#endif

// ---------------------------------------------------------------------------
// Kernel 1: per-row online softmax stats.  One block per (b,s) row.
// Streams 128 KB/row through LDS with double-buffered async tensor loads.
// ---------------------------------------------------------------------------
__global__ __launch_bounds__(256) void row_softmax_stats(
    const float* __restrict__ logits, const int* __restrict__ targets,
    float* __restrict__ nll, float* __restrict__ pnl, float* __restrict__ peos)
{
  const int row = blockIdx.x;
  const int tid = threadIdx.x;
  const float*  lrow  = logits + (size_t)row * VOCAB;
  const float4* lrow4 = (const float4*)lrow;  // 8000 float4 per row

  float m = -INFINITY, s = 0.f;

#if USE_ASYNC
  __shared__ float4 tile[2][1024];  // 2 x 16 KB double buffer
  const int NT = 8;  // 8 tiles of 1024 float4 (last tile: 832 valid)

  // prime tile 0 (4 b128 async loads per thread, uniform per wave)
  #pragma unroll
  for (int k = 0; k < 4; ++k) {
    int j = tid + k * 256;
    int idx4 = j; if (idx4 > 7999) idx4 = 7999;
    __builtin_amdgcn_global_load_async_to_lds_b128(
        (gas_p)(lrow4 + idx4), (las_p)&tile[0][j], 0, 0);
  }
  for (int t = 0; t < NT; ++t) {
    const int cur = t & 1;
    if (t + 1 < NT) {
      const int nxt = (t + 1) & 1;
      #pragma unroll
      for (int k = 0; k < 4; ++k) {
        int j = tid + k * 256;
        int idx4 = (t + 1) * 1024 + j; if (idx4 > 7999) idx4 = 7999;
        __builtin_amdgcn_global_load_async_to_lds_b128(
            (gas_p)(lrow4 + idx4), (las_p)&tile[nxt][j], 0, 0);
      }
      WAIT_ASYNC(4);   // in-order completion: tile t fully landed
    } else {
      WAIT_ASYNC(0);
    }
    __syncthreads();
    const int valid = (t == NT - 1) ? (8000 - t * 1024) : 1024;
    #pragma unroll
    for (int k = 0; k < 4; ++k) {
      int j = tid + k * 256;
      if (j < valid) {
        float4 v = tile[cur][j];
        float m4 = fmaxf(fmaxf(v.x, v.y), fmaxf(v.z, v.w));
        if (m4 > m) { s *= __expf(m - m4); m = m4; }
        s += __expf(v.x - m) + __expf(v.y - m) +
             __expf(v.z - m) + __expf(v.w - m);
      }
    }
    __syncthreads();  // buffer reusable for next async issue
  }
#else
  for (int j = tid; j < 8000; j += 256) {
    float4 v = lrow4[j];
    float m4 = fmaxf(fmaxf(v.x, v.y), fmaxf(v.z, v.w));
    if (m4 > m) { s *= __expf(m - m4); m = m4; }
    s += __expf(v.x - m) + __expf(v.y - m) +
         __expf(v.z - m) + __expf(v.w - m);
  }
#endif

  // wave32 butterfly combine of (m, s)
  for (int off = 16; off > 0; off >>= 1) {
    float om = __shfl_xor(m, off, 32);
    float os = __shfl_xor(s, off, 32);
    float nm = fmaxf(m, om);
    s = s * __expf(m - nm) + os * __expf(om - nm);
    m = nm;
  }
  __shared__ float lm[8], ls[8];
  const int wid = tid >> 5;
  if ((tid & 31) == 0) { lm[wid] = m; ls[wid] = s; }
  __syncthreads();
  if (tid == 0) {
    float M = lm[0], S = ls[0];
    for (int w2 = 1; w2 < 8; ++w2) {
      float om = lm[w2], os = ls[w2];
      float nm = fmaxf(M, om);
      S = S * __expf(M - nm) + os * __expf(om - nm);
      M = nm;
    }
    float logZ = M + __logf(S);
    int   tgt  = targets[row];
    float xt = lrow[tgt];
    float xn = lrow[13];   // NEWLINE
    float xe = lrow[2];    // EOS
    nll[row]  = logZ - xt;
    pnl[row]  = __expf(xn - logZ);
    peos[row] = __expf(xe - logZ);
  }
}

// ---------------------------------------------------------------------------
// Kernel 2: combine per-row stats into the scalar loss.  Single block.
// Final 5 block-sums reduced with V_WMMA_F32_16X16X4_F32 (ones-B row-sum).
// ---------------------------------------------------------------------------
__global__ __launch_bounds__(256) void finalize_loss(
    const float* __restrict__ nll, const float* __restrict__ pnl,
    const float* __restrict__ peos, const int* __restrict__ targets,
    float* __restrict__ out)
{
  const int tid = threadIdx.x;
  float ce_num = 0.f, ce_den = 0.f, len_num = 0.f, mask_sum = 0.f, line_num = 0.f;

  for (int r = tid; r < NROWS; r += 256) {
    int   t    = targets[r];
    float mask = (t != 0) ? 1.f : 0.f;
    float w    = mask * ((t == 13) ? 5.f : 1.f);
    ce_num   += nll[r] * w;
    ce_den   += w;
    mask_sum += mask;

    float pos      = (float)(r & (SEQ - 1));
    float is_eos   = (t == 2) ? 1.f : 0.f;
    float not_eos  = (1.f - is_eos) * mask;
    float too_early = fmaxf(200.f - pos, 0.f);
    float pe = peos[r];
    float contrib = is_eos * too_early * 0.5f;                        // early_eos
    if (pos < 50.f)        contrib += not_eos * pe * 5.f;             // discourage
    else if (pos < 200.f)  contrib += not_eos * pe *
                             ((200.f - pos) * (1.f / (150.f + 1e-8f))) * 2.f; // mid_pen
    len_num += contrib * mask;
  }

  // sequential line-position scan, one lane per batch row
  if (tid < NBATCH) {
    float count = 0.f;
    const int base = tid * SEQ;
    for (int sI = 0; sI < SEQ; ++sI) {
      int  t      = targets[base + sI];
      bool is_nl  = (t == 13);
      bool is_pad = (t == 0);
      float pos = is_nl ? count : (is_pad ? 0.f : count + 1.f);
      count     = is_nl ? 0.f   : (is_pad ? count : count + 1.f);
      float mask = is_pad ? 0.f : 1.f;
      float sp = (is_nl ? 1.f : 0.f) * fmaxf(3.f - pos, 0.f) * 2.f;
      float le = (is_nl ? 0.f : 1.f) * mask * fmaxf(pos - 8.f, 0.f) *
                 (1.f - pnl[base + sI]) * 0.1f;
      line_num += (sp + le) * mask;
    }
  }

  __shared__ float red[5][256];
  red[0][tid] = ce_num;  red[1][tid] = ce_den;  red[2][tid] = mask_sum;
  red[3][tid] = len_num; red[4][tid] = line_num;
  __syncthreads();

  if (tid < 32) {  // full wave0, EXEC all-ones for WMMA
    v2f bones; bones.x = 1.f; bones.y = 1.f;  // B = ones(4x16)
    float totals[5];
    #pragma unroll
    for (int q = 0; q < 5; ++q) {
      v8f c = {};
      #pragma unroll
      for (int ch = 0; ch < 4; ++ch) {
        v2f a;  // A(16x4) f32: row i <- elems {i, i+16, i+32, i+48} of chunk
        a.x = red[q][ch * 64 + tid];
        a.y = red[q][ch * 64 + 32 + tid];
        c = __builtin_amdgcn_wmma_f32_16x16x4_f32(false, a, false, bones,
                                                  (short)0, c, false, false);
      }
      float part = c[0] + c[1] + c[2] + c[3] + c[4] + c[5] + c[6] + c[7];
      part += __shfl_xor(part, 16, 32);  // rows 0-7 half + rows 8-15 half
      totals[q] = part;
    }
    if (tid == 0) {
      float ce   = totals[0] / totals[1];
      float den  = totals[2] + 1e-8f;
      float lenl = totals[3] / den;
      float line = totals[4] / den;
      out[0] = ce + 0.1f * line + 0.1f * lenl;
    }
  }
}

// ---------------------------------------------------------------------------
extern "C" void kernel_launch(void* const* d_in, const int* in_sizes, int n_in,
                              void* d_out, int out_size, void* d_ws, size_t ws_size,
                              hipStream_t stream)
{
  (void)in_sizes; (void)n_in; (void)out_size; (void)ws_size;
  const float* logits  = (const float*)d_in[0];
  const int*   targets = (const int*)d_in[1];
  float* ws   = (float*)d_ws;
  float* nll  = ws;
  float* pnl  = ws + NROWS;
  float* peos = ws + 2 * NROWS;

  row_softmax_stats<<<NROWS, 256, 0, stream>>>(logits, targets, nll, pnl, peos);
  finalize_loss<<<1, 256, 0, stream>>>(nll, pnl, peos, targets, (float*)d_out);
}